// FSQWhisperTokenizer_14010183319976
// MI455X (gfx1250) — compile-verified
//
#include <hip/hip_runtime.h>

typedef __attribute__((ext_vector_type(2))) float v2f;
typedef __attribute__((ext_vector_type(4))) float v4f;
typedef __attribute__((ext_vector_type(8))) float v8f;
typedef __attribute__((ext_vector_type(4))) int   v4i;

typedef __attribute__((address_space(1))) v4i* gptr_v4i;   // global
typedef __attribute__((address_space(3))) v4i* lptr_v4i;   // LDS

#define D_GROUPS 8
#define K_CODES  512
#define SUBDIM   64
#define N_ROWS   32768        // B*T = 16*2048
#define DIM_     512
#define RTILES   16           // row-tiles of 16 rows per block
#define LPITCH   68           // LDS pitch (floats) -> conflict-free frag reads

// ---- CDNA5 async global->LDS copy (ASYNCcnt-tracked, no VGPR round trip) ----
__device__ __forceinline__ void async_tile_load_b128(const float* g, float* l) {
#if __has_builtin(__builtin_amdgcn_global_load_async_to_lds_b128)
  __builtin_amdgcn_global_load_async_to_lds_b128(
      (gptr_v4i)const_cast<float*>(g), (lptr_v4i)l, 0, 0);
#else
  unsigned lofs =
      (unsigned)(size_t)(__attribute__((address_space(3))) void*)l;
  asm volatile("global_load_async_to_lds_b128 %0, %1, off"
               :: "v"(lofs), "v"(g) : "memory");
#endif
}

__device__ __forceinline__ void wait_async0() {
#if __has_builtin(__builtin_amdgcn_s_wait_asynccnt)
  __builtin_amdgcn_s_wait_asynccnt(0);
#else
  asm volatile("s_wait_asynccnt 0x0" ::: "memory");
#endif
}

__global__ __launch_bounds__(256) void vq_wmma_kernel(
    const float* __restrict__ x,
    const float* __restrict__ emb,
    float* __restrict__ out)
{
  __shared__ __align__(16) float ldsA[2][16 * LPITCH];   // double buffer
  __shared__ float sRed[8][16];
  __shared__ int   iRed[8][16];
  __shared__ int   ldsIdx[16];

  const int tid  = threadIdx.x;
  const int wave = tid >> 5;
  const int lane = tid & 31;
  const int l15  = lane & 15;
  const int hi   = lane >> 4;            // which half of the wave

  const int d     = blockIdx.x & (D_GROUPS - 1);
  const int super = blockIdx.x >> 3;

  // ---- load this wave's B fragments (64 codes x 64 dims) into registers ----
  // B layout for V_WMMA_F32_16X16X4_F32: VGPR0 = row K (lanes 0-15) / K+2
  // (lanes 16-31), VGPR1 = K+1 / K+3; N (code) = lane&15.
  v2f  bfr[4][16];
  float en[4];
#pragma unroll
  for (int t = 0; t < 4; ++t) {
    const int code = (wave * 4 + t) * 16 + l15;
    const float* erow = emb + ((size_t)d * K_CODES + code) * SUBDIM;
    float nh = 0.f;
#pragma unroll
    for (int s = 0; s < 16; ++s) {
      v2f b = *(const v2f*)(erow + 4 * s + 2 * hi);
      bfr[t][s] = b;
      nh += b.x * b.x + b.y * b.y;
    }
    // other half of the columns lives in lane^16 -> full ||e_code||^2
    en[t] = nh + __shfl_xor(nh, 16, 32);
  }

  float* out_recon = out;
  float* out_idx   = out + (size_t)N_ROWS * DIM_;
  float* out_loss  = out + (size_t)N_ROWS * DIM_ + (size_t)N_ROWS * D_GROUPS;

  // per-thread staging coordinates (one b128 per lane per tile)
  const int srow = tid >> 4;
  const int sc4  = (tid & 15) * 4;
  const float* gstage = x + (size_t)srow * DIM_ + d * SUBDIM + sc4;
  float* lstage[2] = { &ldsA[0][srow * LPITCH + sc4],
                       &ldsA[1][srow * LPITCH + sc4] };

  // prefetch tile 0 into buffer 0
  async_tile_load_b128(gstage + (size_t)(super * RTILES) * 16 * DIM_,
                       lstage[0]);

  float lossAcc = 0.f;

  for (int r = 0; r < RTILES; ++r) {
    const int cur = r & 1;
    const int n0  = (super * RTILES + r) * 16;

    wait_async0();          // this wave's async writes to ldsA[cur] landed
    __syncthreads();        // everyone's landed; prior readers of nxt done

    if (r + 1 < RTILES)     // stream next tile into the idle buffer
      async_tile_load_b128(gstage + (size_t)(n0 + 16) * DIM_,
                           lstage[cur ^ 1]);

    const float* curBuf = ldsA[cur];

    v8f acc0 = {}, acc1 = {}, acc2 = {}, acc3 = {};
    const float* aBase = curBuf + l15 * LPITCH + 2 * hi;
#pragma unroll
    for (int s = 0; s < 16; ++s) {
      v2f a = *(const v2f*)(aBase + 4 * s);
      acc0 = __builtin_amdgcn_wmma_f32_16x16x4_f32(false, a, false, bfr[0][s],
                                                   (short)0, acc0, false, false);
      acc1 = __builtin_amdgcn_wmma_f32_16x16x4_f32(false, a, false, bfr[1][s],
                                                   (short)0, acc1, false, false);
      acc2 = __builtin_amdgcn_wmma_f32_16x16x4_f32(false, a, false, bfr[2][s],
                                                   (short)0, acc2, false, false);
      acc3 = __builtin_amdgcn_wmma_f32_16x16x4_f32(false, a, false, bfr[3][s],
                                                   (short)0, acc3, false, false);
    }

    // ---- argmin over codes per row ----
    // C layout: VGPR j holds row M=j (lanes 0-15) or M=j+8 (lanes 16-31),
    // code column = ktile*16 + (lane&15).
#pragma unroll
    for (int j = 0; j < 8; ++j) {
      float best = en[0] - 2.f * acc0[j];
      int   bidx = wave * 64 + l15;
      {
        float sc = en[1] - 2.f * acc1[j];
        int   ci = wave * 64 + 16 + l15;
        if (sc < best) { best = sc; bidx = ci; }
      }
      {
        float sc = en[2] - 2.f * acc2[j];
        int   ci = wave * 64 + 32 + l15;
        if (sc < best) { best = sc; bidx = ci; }
      }
      {
        float sc = en[3] - 2.f * acc3[j];
        int   ci = wave * 64 + 48 + l15;
        if (sc < best) { best = sc; bidx = ci; }
      }
#pragma unroll
      for (int off = 8; off >= 1; off >>= 1) {
        float ob = __shfl_xor(best, off, 16);
        int   oi = __shfl_xor(bidx, off, 16);
        if (ob < best || (ob == best && oi < bidx)) { best = ob; bidx = oi; }
      }
      if (l15 == 0) {
        sRed[wave][j + 8 * hi] = best;
        iRed[wave][j + 8 * hi] = bidx;
      }
    }
    __syncthreads();

    if (tid < 16) {
      float best = sRed[0][tid];
      int   bidx = iRed[0][tid];
#pragma unroll
      for (int w = 1; w < 8; ++w) {
        float sc = sRed[w][tid];
        int   ci = iRed[w][tid];
        if (sc < best || (sc == best && ci < bidx)) { best = sc; bidx = ci; }
      }
      ldsIdx[tid] = bidx;
      out_idx[(size_t)(n0 + tid) * D_GROUPS + d] = (float)bidx;
    }
    __syncthreads();

    // ---- gather winning codes, write recon, accumulate loss ----
    {
      const int k = ldsIdx[srow];
      v4f e  = *(const v4f*)(emb + ((size_t)d * K_CODES + k) * SUBDIM + sc4);
      v4f xv = *(const v4f*)(curBuf + srow * LPITCH + sc4);
      *(v4f*)(out_recon + (size_t)(n0 + srow) * DIM_ + d * SUBDIM + sc4) = e;
      v4f df = e - xv;
      lossAcc += df.x * df.x + df.y * df.y + df.z * df.z + df.w * df.w;
    }
  }

  // ---- loss: wave reduce + one atomic per wave ----
#pragma unroll
  for (int off = 16; off >= 1; off >>= 1)
    lossAcc += __shfl_xor(lossAcc, off, 32);
  if (lane == 0)
    atomicAdd(out_loss, lossAcc * (1.25f / ((float)N_ROWS * (float)DIM_)));
}

__global__ void vq_zero_loss(float* p) { *p = 0.f; }

extern "C" void kernel_launch(void* const* d_in, const int* in_sizes, int n_in,
                              void* d_out, int out_size, void* d_ws, size_t ws_size,
                              hipStream_t stream) {
  const float* x   = (const float*)d_in[0];   // (16, 2048, 512) f32
  const float* emb = (const float*)d_in[1];   // (8, 512, 64)   f32
  float* out = (float*)d_out;                 // recon | indices | loss

  vq_zero_loss<<<1, 1, 0, stream>>>(
      out + (size_t)N_ROWS * DIM_ + (size_t)N_ROWS * D_GROUPS);

  const int blocks = (N_ROWS / (16 * RTILES)) * D_GROUPS;  // 128 * 8 = 1024
  vq_wmma_kernel<<<blocks, 256, 0, stream>>>(x, emb, out);
}